// Attack_net_withlabelsLP_clamped_67654324847190
// MI455X (gfx1250) — compile-verified
//
#include <hip/hip_runtime.h>

typedef float v2f __attribute__((ext_vector_type(2)));
typedef float v8f __attribute__((ext_vector_type(8)));

#define C_CLASSES 10
#define LRELU(v) ((v) > 0.0f ? (v) : 0.01f * (v))

// ---------------- setup kernels: counting sort by class ----------------

__global__ void k_zero(int* __restrict__ p) {
  int i = threadIdx.x;
  if (i < 64) p[i] = 0;
}

__global__ void k_hist(const int* __restrict__ y, int* __restrict__ hist, int n) {
  int i = blockIdx.x * blockDim.x + threadIdx.x;
  if (i < n) atomicAdd(&hist[y[i]], 1);
}

__global__ void k_scan_tiles(const int* __restrict__ hist, int* __restrict__ offs,
                             int* __restrict__ numTiles, int3* __restrict__ tiles) {
  __shared__ int s_off[C_CLASSES], s_cnt[C_CLASSES], s_tb[C_CLASSES];
  if (threadIdx.x == 0) {
    int off = 0, tb = 0;
    for (int c = 0; c < C_CLASSES; ++c) {
      s_cnt[c] = hist[c];
      s_off[c] = off;
      offs[c] = off;
      s_tb[c] = tb;
      off += s_cnt[c];
      tb += (s_cnt[c] + 15) >> 4;
    }
    *numTiles = tb;
  }
  __syncthreads();
  for (int c = 0; c < C_CLASSES; ++c) {
    int nt = (s_cnt[c] + 15) >> 4;
    for (int t = threadIdx.x; t < nt; t += blockDim.x) {
      int st = s_off[c] + t * 16;
      int len = s_cnt[c] - t * 16;
      if (len > 16) len = 16;
      tiles[s_tb[c] + t] = make_int3(c, st, len);
    }
  }
}

__global__ void k_scatter(const int* __restrict__ y, const int* __restrict__ offs,
                          int* __restrict__ cursor, int* __restrict__ sortedIdx, int n) {
  int i = blockIdx.x * blockDim.x + threadIdx.x;
  if (i < n) {
    int c = y[i];
    int p = offs[c] + atomicAdd(&cursor[c], 1);
    sortedIdx[p] = i;
  }
}

// ---------------- weight pre-pack: B fragments in per-lane WMMA layout ----------------
// packed[(c*numNt*numKs + nt*numKs + ks)*64 + lane*2 + comp] =
//   W[c][kr][col], kr = ks*4 + (lane>>4)*2 + comp, col = nt*16 + (lane&15), zero-padded.

__global__ void k_pack_b(const float* __restrict__ W, float* __restrict__ outp,
                         int K, int Ncols, int numKs, int numNt, int nCls) {
  int i = blockIdx.x * blockDim.x + threadIdx.x;
  int perCls = numNt * numKs * 64;
  int total = perCls * nCls;
  if (i >= total) return;
  int c = i / perCls;
  int r = i % perCls;
  int comp = r & 1;
  int lane = (r >> 1) & 31;
  int step = r >> 6;  // nt*numKs + ks
  int ks = step % numKs;
  int nt = step / numKs;
  int row = lane & 15, hi = lane >> 4;
  int col = nt * 16 + row;
  int kr = ks * 4 + hi * 2 + comp;
  outp[i] = (kr < K && col < Ncols) ? W[c * K * Ncols + kr * Ncols + col] : 0.0f;
}

__global__ void k_pack_bias(const float* __restrict__ b, float* __restrict__ outp,
                            int Ncols, int Npad, int nCls) {
  int i = blockIdx.x * blockDim.x + threadIdx.x;
  int total = Npad * nCls;
  if (i >= total) return;
  int c = i / Npad, col = i % Npad;
  outp[i] = (col < Ncols) ? b[c * Ncols + col] : 0.0f;
}

// ---------------- main routed MLP kernel: 1 wave32 per 16-sample tile ----------------

__global__ __launch_bounds__(32) void k_mlp_route(
    const float* __restrict__ x,
    const float* __restrict__ We1, const float* __restrict__ be1,
    const float* __restrict__ pWe2, const float* __restrict__ pbe2,
    const float* __restrict__ pWd1, const float* __restrict__ pbd1,
    const float* __restrict__ pWd2, const float* __restrict__ pbd2,
    const float* __restrict__ pWd3, const float* __restrict__ pbd3,
    const int* __restrict__ sortedIdx, const int3* __restrict__ tiles,
    const int* __restrict__ numTiles, float* __restrict__ out) {
  // LDS strides chosen so 16 rows hit 16 distinct bank groups (stride % 64 in {4,20,52})
  __shared__ float sh1[16 * 52];   // h1: 16x50 (+2 zero pad)
  __shared__ float sh2[16 * 116];  // h2: 16x100
  __shared__ float sa1[16 * 68];   // a1: 16x50 (cols 50..63 zero)
  __shared__ float sa2[16 * 20];   // a2: 16x15 (col 15 zero)
  __shared__ float so[16 * 20];    // o : 16x2

  const int t = blockIdx.x;
  if (t >= *numTiles) return;  // wave-uniform
  const int3 td = tiles[t];
  const int cls = td.x, start = td.y, len = td.z;

  const int lane = threadIdx.x;
  const int row = lane & 15;  // M row of tile / N col of B fragment
  const int hi = lane >> 4;
  const int koff = hi * 2;
  const int l2 = lane * 2;

  __builtin_prefetch(pWd1 + cls * 4 * 25 * 64, 0, 0);  // global_prefetch_b8
  __builtin_prefetch(pWe2, 0, 0);

  const bool valid = (row < len);
  const int sIdx = sortedIdx[start + (valid ? row : 0)];
  const float x0 = valid ? x[2 * sIdx + 0] : 0.0f;
  const float x1 = valid ? x[2 * sIdx + 1] : 0.0f;

  // ---- encoder layer 1 (K=2, VALU): h1 = lrelu(x @ We1 + be1), 16x50
  #pragma unroll
  for (int jj = 0; jj < 25; ++jj) {
    int j = hi * 25 + jj;
    float v = fmaf(x0, We1[j], fmaf(x1, We1[50 + j], be1[j]));
    sh1[row * 52 + j] = LRELU(v);
  }
  if (hi == 0) { sh1[row * 52 + 50] = 0.0f; sh1[row * 52 + 51] = 0.0f; }
  __syncthreads();

  // ---- encoder layer 2 + lrelu: h2 = lrelu(h1 @ We2 + be2), 16x100 (7 nt, 13 ks)
  {
    v2f af[13];
    #pragma unroll
    for (int ks = 0; ks < 13; ++ks) {
      af[ks].x = sh1[row * 52 + ks * 4 + koff];
      af[ks].y = sh1[row * 52 + ks * 4 + koff + 1];
    }
    for (int nt = 0; nt < 7; ++nt) {
      const float* bp = pWe2 + nt * 13 * 64 + l2;
      v8f acc0 = {0.f, 0.f, 0.f, 0.f, 0.f, 0.f, 0.f, 0.f};
      v8f acc1 = {0.f, 0.f, 0.f, 0.f, 0.f, 0.f, 0.f, 0.f};
      #pragma unroll
      for (int ks = 0; ks < 13; ++ks) {
        v2f b = *(const v2f*)(bp + ks * 64);
        if (ks & 1)
          acc1 = __builtin_amdgcn_wmma_f32_16x16x4_f32(false, af[ks], false, b,
                                                       (short)0, acc1, false, false);
        else
          acc0 = __builtin_amdgcn_wmma_f32_16x16x4_f32(false, af[ks], false, b,
                                                       (short)0, acc0, false, false);
      }
      const int col = nt * 16 + row;
      const float bias = pbe2[col];
      #pragma unroll
      for (int r = 0; r < 8; ++r) {
        float v = acc0[r] + acc1[r] + bias;
        sh2[(r + hi * 8) * 116 + col] = LRELU(v);
      }
    }
  }
  __syncthreads();

  // ---- decoder layer 1 + lrelu: a1 = lrelu(h2 @ Wd1[c] + bd1[c]), 16x50 (4 nt, 25 ks)
  {
    v2f af[25];
    #pragma unroll
    for (int ks = 0; ks < 25; ++ks) {
      af[ks].x = sh2[row * 116 + ks * 4 + koff];
      af[ks].y = sh2[row * 116 + ks * 4 + koff + 1];
    }
    for (int nt = 0; nt < 4; ++nt) {
      const float* bp = pWd1 + (cls * 4 + nt) * 25 * 64 + l2;
      v8f acc0 = {0.f, 0.f, 0.f, 0.f, 0.f, 0.f, 0.f, 0.f};
      v8f acc1 = {0.f, 0.f, 0.f, 0.f, 0.f, 0.f, 0.f, 0.f};
      #pragma unroll
      for (int ks = 0; ks < 25; ++ks) {
        v2f b = *(const v2f*)(bp + ks * 64);
        if (ks & 1)
          acc1 = __builtin_amdgcn_wmma_f32_16x16x4_f32(false, af[ks], false, b,
                                                       (short)0, acc1, false, false);
        else
          acc0 = __builtin_amdgcn_wmma_f32_16x16x4_f32(false, af[ks], false, b,
                                                       (short)0, acc0, false, false);
      }
      const int col = nt * 16 + row;
      const float bias = pbd1[cls * 64 + col];
      #pragma unroll
      for (int r = 0; r < 8; ++r) {
        float v = acc0[r] + acc1[r] + bias;
        sa1[(r + hi * 8) * 68 + col] = LRELU(v);
      }
    }
  }
  __syncthreads();

  // ---- decoder layer 2 + lrelu: a2 = lrelu(a1 @ Wd2[c] + bd2[c]), 16x15 (1 nt, 13 ks)
  {
    const float* bp = pWd2 + cls * 13 * 64 + l2;
    v8f acc0 = {0.f, 0.f, 0.f, 0.f, 0.f, 0.f, 0.f, 0.f};
    v8f acc1 = {0.f, 0.f, 0.f, 0.f, 0.f, 0.f, 0.f, 0.f};
    #pragma unroll
    for (int ks = 0; ks < 13; ++ks) {
      v2f a;
      a.x = sa1[row * 68 + ks * 4 + koff];
      a.y = sa1[row * 68 + ks * 4 + koff + 1];
      v2f b = *(const v2f*)(bp + ks * 64);
      if (ks & 1)
        acc1 = __builtin_amdgcn_wmma_f32_16x16x4_f32(false, a, false, b, (short)0,
                                                     acc1, false, false);
      else
        acc0 = __builtin_amdgcn_wmma_f32_16x16x4_f32(false, a, false, b, (short)0,
                                                     acc0, false, false);
    }
    const float bias = pbd2[cls * 16 + row];
    #pragma unroll
    for (int r = 0; r < 8; ++r) {
      float v = acc0[r] + acc1[r] + bias;
      sa2[(r + hi * 8) * 20 + row] = LRELU(v);
    }
  }
  __syncthreads();

  // ---- decoder layer 3 (no activation): o = a2 @ Wd3[c] + bd3[c], 16x2 (4 ks)
  {
    const float* bp = pWd3 + cls * 4 * 64 + l2;
    v8f acc = {0.f, 0.f, 0.f, 0.f, 0.f, 0.f, 0.f, 0.f};
    #pragma unroll
    for (int ks = 0; ks < 4; ++ks) {
      v2f a;
      a.x = sa2[row * 20 + ks * 4 + koff];
      a.y = sa2[row * 20 + ks * 4 + koff + 1];
      v2f b = *(const v2f*)(bp + ks * 64);
      acc = __builtin_amdgcn_wmma_f32_16x16x4_f32(false, a, false, b, (short)0, acc,
                                                  false, false);
    }
    const float bias = pbd3[cls * 16 + row];
    #pragma unroll
    for (int r = 0; r < 8; ++r) {
      so[(r + hi * 8) * 20 + row] = acc[r] + bias;
    }
  }
  __syncthreads();

  // ---- finalize: L2-normalize, scale by sqrt(2)*eps, clamp, scatter to out
  if (hi == 0 && valid) {
    const float o0 = so[row * 20 + 0];
    const float o1 = so[row * 20 + 1];
    const float nrm = sqrtf(fmaf(o0, o0, o1 * o1));
    const float s = 0.70710678118654752f / nrm;  // D^(1/P) * EPS
    float r0 = o0 * s, r1 = o1 * s;
    r0 = fminf(fmaxf(r0, -0.5f), 0.5f);
    r1 = fminf(fmaxf(r1, -0.5f), 0.5f);
    out[2 * sIdx + 0] = r0;
    out[2 * sIdx + 1] = r1;
  }
}

// ---------------- host launcher ----------------

extern "C" void kernel_launch(void* const* d_in, const int* in_sizes, int n_in,
                              void* d_out, int out_size, void* d_ws, size_t ws_size,
                              hipStream_t stream) {
  const float* x   = (const float*)d_in[0];
  const int*   y   = (const int*)d_in[1];
  const float* We1 = (const float*)d_in[2];
  const float* be1 = (const float*)d_in[3];
  const float* We2 = (const float*)d_in[4];
  const float* be2 = (const float*)d_in[5];
  const float* Wd1 = (const float*)d_in[6];
  const float* bd1 = (const float*)d_in[7];
  const float* Wd2 = (const float*)d_in[8];
  const float* bd2 = (const float*)d_in[9];
  const float* Wd3 = (const float*)d_in[10];
  const float* bd3 = (const float*)d_in[11];
  float* out = (float*)d_out;

  const int n = in_sizes[0] / 2;  // N samples (x is [N,2])
  const int maxTiles = (n + 15) / 16 + C_CLASSES;

  // ---- workspace layout: packed floats first (8B-aligned), then int region
  float* wsf = (float*)d_ws;
  float* pWe2 = wsf;               // 7*13*64   = 5824
  float* pWd1 = pWe2 + 5824;       // 10*4*25*64= 64000
  float* pWd2 = pWd1 + 64000;      // 10*13*64  = 8320
  float* pWd3 = pWd2 + 8320;       // 10*4*64   = 2560
  float* pbe2 = pWd3 + 2560;       // 112 (pad to 128)
  float* pbd1 = pbe2 + 128;        // 10*64 = 640
  float* pbd2 = pbd1 + 640;        // 10*16 = 160
  float* pbd3 = pbd2 + 160;        // 10*16 = 160
  int* wsi = (int*)(pbd3 + 160);
  int* hist = wsi + 0;
  int* cursor = wsi + 16;
  int* offs = wsi + 32;
  int* numTiles = wsi + 48;
  int3* tiles = (int3*)(wsi + 64);
  int* sortedIdx = wsi + 64 + 3 * maxTiles;

  const int tb = 256;
  const int nblk = (n + tb - 1) / tb;
  #define GRID(total) (((total) + tb - 1) / tb)

  // sort pipeline
  k_zero<<<1, 64, 0, stream>>>(wsi);
  k_hist<<<nblk, tb, 0, stream>>>(y, hist, n);
  k_scan_tiles<<<1, 256, 0, stream>>>(hist, offs, numTiles, tiles);
  k_scatter<<<nblk, tb, 0, stream>>>(y, offs, cursor, sortedIdx, n);

  // weight packing (cheap; keeps main-kernel inner loop branch-free)
  k_pack_b<<<GRID(5824), tb, 0, stream>>>(We2, pWe2, 50, 100, 13, 7, 1);
  k_pack_b<<<GRID(64000), tb, 0, stream>>>(Wd1, pWd1, 100, 50, 25, 4, C_CLASSES);
  k_pack_b<<<GRID(8320), tb, 0, stream>>>(Wd2, pWd2, 50, 15, 13, 1, C_CLASSES);
  k_pack_b<<<GRID(2560), tb, 0, stream>>>(Wd3, pWd3, 15, 2, 4, 1, C_CLASSES);
  k_pack_bias<<<GRID(112), tb, 0, stream>>>(be2, pbe2, 100, 112, 1);
  k_pack_bias<<<GRID(640), tb, 0, stream>>>(bd1, pbd1, 50, 64, C_CLASSES);
  k_pack_bias<<<GRID(160), tb, 0, stream>>>(bd2, pbd2, 15, 16, C_CLASSES);
  k_pack_bias<<<GRID(160), tb, 0, stream>>>(bd3, pbd3, 2, 16, C_CLASSES);

  k_mlp_route<<<maxTiles, 32, 0, stream>>>(x, We1, be1, pWe2, pbe2, pWd1, pbd1, pWd2,
                                           pbd2, pWd3, pbd3, sortedIdx, tiles,
                                           numTiles, out);
  #undef GRID
}